// Upsampling_68882685493276
// MI455X (gfx1250) — compile-verified
//
#include <hip/hip_runtime.h>

#ifndef __has_builtin
#define __has_builtin(x) 0
#endif

#if __has_builtin(__builtin_amdgcn_global_load_async_to_lds_b128) && \
    __has_builtin(__builtin_amdgcn_s_wait_asynccnt)
#define USE_ASYNC_LDS 1
#else
#define USE_ASYNC_LDS 0
#endif

typedef float f32x4 __attribute__((ext_vector_type(4)));
typedef int   v4i   __attribute__((ext_vector_type(4)));

// Builtin signature (probe-confirmed round 1->2): pointers to 128-bit int
// vectors, global side addrspace(1), LDS side addrspace(3).
typedef __attribute__((address_space(1))) v4i gv4i_t;
typedef __attribute__((address_space(3))) v4i lv4i_t;

// Fixed problem shape: x (16,64,256,256) f32 -> out (16,64,512,512) f32.
// 2x nearest-neighbor upsample: out[bc, 2h+{0,1}, 2w+{0,1}] = x[bc, h, w].

__device__ __forceinline__ void store_dup2x_at(float* p /*row 2h, col 2w*/, f32x4 v) {
    f32x4 lo = {v.x, v.x, v.y, v.y};
    f32x4 hi = {v.z, v.z, v.w, v.w};
    f32x4* p0 = (f32x4*)p;            // row 2h
    f32x4* p1 = (f32x4*)(p + 512);    // row 2h+1
    // 1.07 GB output streams once (>> 192MB L2) -> non-temporal stores.
    __builtin_nontemporal_store(lo, p0);
    __builtin_nontemporal_store(hi, p0 + 1);
    __builtin_nontemporal_store(lo, p1);
    __builtin_nontemporal_store(hi, p1 + 1);
}

#if USE_ASYNC_LDS
constexpr int TPB           = 256;  // 8 wave32s
constexpr int TILES_PER_BLK = 16;   // 16 tiles * 1024 floats = 64 input rows / block
constexpr int NBUF          = 4;    // 4 x 4KB LDS ring, 3 tiles in flight

// Block-contiguous async streaming pipeline.
// Block blk owns float4s [blk*4096, (blk+1)*4096): exactly a quarter of one
// (b,c) image (64 of 256 rows), so bc / h0 / w4 are loop-invariant and both
// stream pointers advance by compile-time constants each iteration.
// Each thread consumes only the LDS slot it issued; async loads complete
// in order, so s_wait_asynccnt alone synchronizes -- no barriers.
__global__ __launch_bounds__(TPB)
void upsample2x_async(const float* __restrict__ x, float* __restrict__ out) {
    __shared__ float buf[NBUF][TPB * 4];
    const int  tid = threadIdx.x;
    const long blk = blockIdx.x;

    const long g0 = blk * (long)(TILES_PER_BLK * TPB);      // first float4 index
    const float* in_p = x + (g0 + tid) * 4;                 // +1024 floats / iter

    const long bc = blk >> 2;                               // 4 blocks per image
    const long h0 = (blk & 3) * 64 + (tid >> 6);            // input row at iter 0
    const long w4 = tid & 63;                               // float4 column
    float* out_p = out + bc * 262144L + h0 * 1024L + w4 * 8L;  // +4096 / iter

    // Prologue: put NBUF-1 = 3 tiles in flight.
    #pragma unroll
    for (int k = 0; k < NBUF - 1; ++k)
        __builtin_amdgcn_global_load_async_to_lds_b128(
            (gv4i_t*)(in_p + k * (TPB * 4)),
            (lv4i_t*)&buf[k][tid * 4], 0, 0);

    #pragma unroll
    for (int i = 0; i < TILES_PER_BLK; ++i) {
        if (i + (NBUF - 1) < TILES_PER_BLK) {
            __builtin_amdgcn_global_load_async_to_lds_b128(
                (gv4i_t*)(in_p + (i + NBUF - 1) * (TPB * 4)),
                (lv4i_t*)&buf[(i + NBUF - 1) % NBUF][tid * 4], 0, 0);
            __builtin_amdgcn_s_wait_asynccnt(3);   // tile i landed (in-order)
        } else if (i == TILES_PER_BLK - 3) {
            __builtin_amdgcn_s_wait_asynccnt(2);
        } else if (i == TILES_PER_BLK - 2) {
            __builtin_amdgcn_s_wait_asynccnt(1);
        } else {                                   // i == TILES_PER_BLK - 1
            __builtin_amdgcn_s_wait_asynccnt(0);
        }
        f32x4 v = *(const f32x4*)&buf[i % NBUF][tid * 4];   // ds_load_b128
        store_dup2x_at(out_p + (long)i * 4096, v);
    }
}
#endif

// Fallback / tail-safe direct streaming kernel (handles arbitrary n4):
// NT B128 load -> 4x NT B128 store, with global_prefetch one stride ahead.
__global__ __launch_bounds__(256)
void upsample2x_direct(const float* __restrict__ x, float* __restrict__ out, long n4) {
    long g = (long)blockIdx.x * blockDim.x + threadIdx.x;
    const long stride = (long)gridDim.x * blockDim.x;
    const f32x4* __restrict__ x4 = (const f32x4*)x;
    for (; g < n4; g += stride) {
        __builtin_prefetch(x4 + g + stride, 0, 0);   // global_prefetch_b8
        f32x4 v = __builtin_nontemporal_load(x4 + g);
        long w4 = g & 63, row = g >> 6;
        long h = row & 255, bc = row >> 8;
        store_dup2x_at(out + bc * 262144L + h * 1024L + w4 * 8L, v);
    }
}

extern "C" void kernel_launch(void* const* d_in, const int* in_sizes, int n_in,
                              void* d_out, int out_size, void* d_ws, size_t ws_size,
                              hipStream_t stream) {
    const float* x = (const float*)d_in[0];
    float* out = (float*)d_out;
    long n  = in_sizes[0];   // 16*64*256*256 = 2^26
    long n4 = n >> 2;        // 2^24 float4s

#if USE_ASYNC_LDS
    long nblk = n4 / (TILES_PER_BLK * TPB);            // 4096 blocks for this shape
    if (nblk > 0)
        upsample2x_async<<<(int)nblk, TPB, 0, stream>>>(x, out);
    long done4 = nblk * (TILES_PER_BLK * TPB);
    if (done4 < n4)                                    // generic tail (unused here)
        upsample2x_direct<<<64, 256, 0, stream>>>(x + done4 * 4, out, n4 - done4);
#else
    upsample2x_direct<<<8192, 256, 0, stream>>>(x, out, n4);
#endif
}